// GCNMultiRegressor_58265526337787
// MI455X (gfx1250) — compile-verified
//
#include <hip/hip_runtime.h>
#include <hip/hip_bf16.h>

typedef __attribute__((ext_vector_type(2))) float v2f;
typedef __attribute__((ext_vector_type(8))) float v8f;

#define N_NODES 50000
#define N_EDGES 800000
#define FEATS   96
#define FQ      (FEATS / 4)   // 24 float4 per row
#define OUTF    8

// ---------------- zero fill ----------------
__global__ void zero_kernel(float* __restrict__ p, unsigned n) {
    unsigned i = blockIdx.x * blockDim.x + threadIdx.x;
    if (i < n) p[i] = 0.0f;
}

__global__ void zero4_kernel(float4* __restrict__ p, unsigned n4) {
    unsigned i = blockIdx.x * blockDim.x + threadIdx.x;
    if (i < n4) p[i] = make_float4(0.f, 0.f, 0.f, 0.f);
}

// ---------------- degrees ----------------
__global__ void degree_kernel(const int* __restrict__ src, const int* __restrict__ dst,
                              float* __restrict__ degO, float* __restrict__ degI, int E) {
    int e = blockIdx.x * blockDim.x + threadIdx.x;
    if (e < E) {
        atomicAdd(&degO[src[e]], 1.0f);
        atomicAdd(&degI[dst[e]], 1.0f);
    }
}

// deg -> 1/sqrt(max(deg,1)) in place, both arrays
__global__ void norm_kernel(float* __restrict__ degO, float* __restrict__ degI, int n) {
    int i = blockIdx.x * blockDim.x + threadIdx.x;
    if (i < n) {
        degO[i] = 1.0f / sqrtf(fmaxf(degO[i], 1.0f));
        degI[i] = 1.0f / sqrtf(fmaxf(degI[i], 1.0f));
    }
}

// ---------------- edge scatter-add: agg[dst] += x[src]*norm_out[src] ----------------
// One thread per (edge, float4-chunk): b128 feature loads, 4 coalesced f32 atomics.
__global__ __launch_bounds__(256) void scatter_kernel(const float4* __restrict__ x4,
                                                      const int* __restrict__ src,
                                                      const int* __restrict__ dst,
                                                      const float* __restrict__ norm_out,
                                                      float* __restrict__ agg,
                                                      unsigned total /* E * FQ */) {
    unsigned t = blockIdx.x * blockDim.x + threadIdx.x;
    if (t >= total) return;
    unsigned e = t / FQ;
    unsigned q = t - e * FQ;
    int s = src[e];
    int d = dst[e];
    float ns = norm_out[s];
    float4 v = x4[(unsigned)s * FQ + q];
    float* p = agg + (unsigned)d * FEATS + q * 4;
    atomicAdd(p + 0, v.x * ns);
    atomicAdd(p + 1, v.y * ns);
    atomicAdd(p + 2, v.z * ns);
    atomicAdd(p + 3, v.w * ns);
}

// ---------------- WMMA GEMM: out = relu?( (agg * norm_in[:,None]) @ W + b ) ----------------
// One wave computes a full 16x96 output strip: A panel (scaled by norm_in) is
// loaded once per K-step and fed to 6 V_WMMA_F32_16X16X4_F32 accumulators.
__global__ __launch_bounds__(128) void gemm_wmma_kernel(const float* __restrict__ A,   // [N,96] agg
                                                        const float* __restrict__ norm_in,
                                                        const float* __restrict__ W,   // [96,96] (K,N)
                                                        const float* __restrict__ bias,// [96]
                                                        float* __restrict__ out,       // [N,96]
                                                        int nrows, int do_relu) {
    const int NT = FEATS / 16;                       // 6 tiles along N
    int wave = blockIdx.x * (blockDim.x >> 5) + (threadIdx.x >> 5);
    int m0 = wave * 16;
    if (m0 >= nrows) return;                         // uniform per wave

    int lane = threadIdx.x & 31;
    int half = lane >> 4;                            // 0: K pair {0,1}, 1: {2,3}
    int l    = lane & 15;                            // row (A) / col (B,D)

    const float* arow = A + (unsigned)(m0 + l) * FEATS;
    float nscale = norm_in[m0 + l];

    v8f c[NT];
    #pragma unroll
    for (int nt = 0; nt < NT; ++nt)
        c[nt] = (v8f){0.f, 0.f, 0.f, 0.f, 0.f, 0.f, 0.f, 0.f};

    for (int k0 = 0; k0 < FEATS; k0 += 4) {
        int ka = k0 + half * 2;
        // A 16x4 fp32 layout: lanes 0-15 hold K=0,1 in v[0],v[1]; lanes 16-31 K=2,3
        v2f a;
        a.x = arow[ka]     * nscale;
        a.y = arow[ka + 1] * nscale;
        const float* w0 = W + (unsigned)ka * FEATS + l;
        #pragma unroll
        for (int nt = 0; nt < NT; ++nt) {
            // B 4x16 fp32 layout mirrors A: lanes 0-15 rows K=0,1; lanes 16-31 rows K=2,3
            v2f b;
            b.x = w0[nt * 16];
            b.y = w0[nt * 16 + FEATS];
            c[nt] = __builtin_amdgcn_wmma_f32_16x16x4_f32(false, a, false, b, (short)0,
                                                          c[nt], false, false);
        }
    }

    // D layout: VGPR r -> row m0 + r + half*8, col n0 + l
    #pragma unroll
    for (int nt = 0; nt < NT; ++nt) {
        float bv = bias[nt * 16 + l];
        #pragma unroll
        for (int r = 0; r < 8; ++r) {
            float v = c[nt][r] + bv;
            if (do_relu) v = fmaxf(v, 0.0f);
            out[(unsigned)(m0 + r + half * 8) * FEATS + nt * 16 + l] = v;
        }
    }
}

// ---------------- column sums over rows ----------------
__global__ void colsum_kernel(const float* __restrict__ x, float* __restrict__ colsum, int nrows) {
    int f = threadIdx.x;
    if (f < FEATS) {
        float s = 0.0f;
        for (int r = blockIdx.x; r < nrows; r += gridDim.x)
            s += x[(unsigned)r * FEATS + f];
        atomicAdd(&colsum[f], s);
    }
}

// ---------------- readout: out = (colsum/N) @ Wr.T + br ----------------
__global__ void readout_kernel(const float* __restrict__ colsum,
                               const float* __restrict__ Wr,  // [8,96]
                               const float* __restrict__ br,  // [8]
                               float* __restrict__ out, float invN) {
    int o = threadIdx.x;
    if (o < OUTF) {
        float s = 0.0f;
        for (int f = 0; f < FEATS; ++f) s += colsum[f] * Wr[o * FEATS + f];
        out[o] = s * invN + br[o];
    }
}

extern "C" void kernel_launch(void* const* d_in, const int* in_sizes, int n_in,
                              void* d_out, int out_size, void* d_ws, size_t ws_size,
                              hipStream_t stream) {
    const float* features = (const float*)d_in[0];
    const int*   src      = (const int*)d_in[1];
    const int*   dst      = (const int*)d_in[2];
    const float* W1       = (const float*)d_in[3];
    const float* b1       = (const float*)d_in[4];
    const float* W2       = (const float*)d_in[5];
    const float* b2       = (const float*)d_in[6];
    const float* Wr       = (const float*)d_in[7];
    const float* br       = (const float*)d_in[8];
    float* out = (float*)d_out;

    // workspace layout (floats); all sub-buffers stay 16B-aligned
    float* ws = (float*)d_ws;
    float* norm_out = ws;                             // 50000 (deg_out -> norm_out in place)
    float* norm_in  = norm_out + N_NODES;             // 50000
    float* bufA     = norm_in + N_NODES;              // 50000*96  (agg)
    float* bufB     = bufA + (size_t)N_NODES * FEATS; // 50000*96  (hidden / output feats)
    float* colsum   = bufB + (size_t)N_NODES * FEATS; // 96

    const unsigned NF = (unsigned)N_NODES * FEATS;    // 4.8M floats
    const unsigned EQ = (unsigned)N_EDGES * FQ;       // 19.2M (edge, float4) work items

    // 1) zero degrees + agg + colsum
    zero_kernel<<<(2 * N_NODES + 255) / 256, 256, 0, stream>>>(norm_out, 2 * N_NODES);
    zero4_kernel<<<(NF / 4 + 255) / 256, 256, 0, stream>>>((float4*)bufA, NF / 4);
    zero_kernel<<<1, 128, 0, stream>>>(colsum, FEATS);

    // 2) degrees -> norms
    degree_kernel<<<(N_EDGES + 255) / 256, 256, 0, stream>>>(src, dst, norm_out, norm_in, N_EDGES);
    norm_kernel<<<(N_NODES + 255) / 256, 256, 0, stream>>>(norm_out, norm_in, N_NODES);

    const int gemm_waves  = N_NODES / 16;             // 3125 strips of 16 rows
    const int gemm_blocks = (gemm_waves + 3) / 4;     // 4 waves / 128-thread block

    // 3) layer 1: scatter features -> bufA, GEMM(bufA*norm_in, W1)+b1, relu -> bufB
    scatter_kernel<<<(EQ + 255) / 256, 256, 0, stream>>>((const float4*)features, src, dst,
                                                         norm_out, bufA, EQ);
    gemm_wmma_kernel<<<gemm_blocks, 128, 0, stream>>>(bufA, norm_in, W1, b1, bufB, N_NODES, 1);

    // 4) layer 2: re-zero agg, scatter bufB -> bufA, GEMM -> bufB (safe: gemm reads only
    //    bufA/W2 and overwrites every element of bufB)
    zero4_kernel<<<(NF / 4 + 255) / 256, 256, 0, stream>>>((float4*)bufA, NF / 4);
    scatter_kernel<<<(EQ + 255) / 256, 256, 0, stream>>>((const float4*)bufB, src, dst,
                                                         norm_out, bufA, EQ);
    gemm_wmma_kernel<<<gemm_blocks, 128, 0, stream>>>(bufA, norm_in, W2, b2, bufB, N_NODES, 0);

    // 5) mean-pool + readout
    colsum_kernel<<<480, 128, 0, stream>>>(bufB, colsum, N_NODES);
    readout_kernel<<<1, 32, 0, stream>>>(colsum, Wr, br, out, 1.0f / (float)N_NODES);
}